// GAT_45913200395030
// MI455X (gfx1250) — compile-verified
//
#include <hip/hip_runtime.h>
#include <hip/hip_bf16.h>
#include <math.h>

typedef __attribute__((ext_vector_type(2))) float v2f;
typedef __attribute__((ext_vector_type(8))) float v8f;

#define N_NODES 10000
#define N_EDGES 320000
#define F1      128   // input feature dim (= K for both GEMMs)
#define H1      4     // heads, layer 1
#define D1      128   // out_dim, layer 1
#define HD1     512   // H1*D1 (= N of GEMM1)
#define C2      22    // out_dim, layer 2

// ---- monotonic float<->uint key for atomicMax-based segment max ----
__device__ __forceinline__ unsigned fkey(float f) {
    unsigned b = __float_as_uint(f);
    return (b & 0x80000000u) ? ~b : (b | 0x80000000u);
}
__device__ __forceinline__ float fdec(unsigned k) {
    unsigned b = (k & 0x80000000u) ? (k ^ 0x80000000u) : ~k;
    return __uint_as_float(b);
}

// ---------------------------------------------------------------------------
// GEMM1 specialized: D(10000x512) = A(10000x128) @ B(128x512), row-major f32.
// One wave computes a 16x64 strip: 4 adjacent 16x16 N-tiles sharing one A
// fragment per k-step (4 back-to-back v_wmma_f32_16x16x4_f32). No guards.
// A frag: lanes 0-15 -> M=lane, K={k0,k0+1}; lanes 16-31 -> K={k0+2,k0+3}.
// B frag mirrored over N. C/D: VGPR v -> row v + 8*(lane>=16), col lane%16.
// ---------------------------------------------------------------------------
__global__ void wmma_gemm1_kernel(const float* __restrict__ A,
                                  const float* __restrict__ B,
                                  float* __restrict__ D) {
    const int wid  = (blockIdx.x * blockDim.x + threadIdx.x) >> 5;
    const int lane = threadIdx.x & 31;
    const int tM   = wid >> 3;        // 625 M-tiles
    const int tN4  = wid & 7;         // 8 strips of 64 columns
    const int half = lane >> 4;
    const int ln   = lane & 15;
    const int m    = tM * 16 + ln;
    const int nb   = tN4 * 64 + ln;   // base column owned by this lane

    v8f acc0 = {}, acc1 = {}, acc2 = {}, acc3 = {};
    const float* arow = A + (size_t)m * F1;

    #pragma unroll 4
    for (int k0 = 0; k0 < F1; k0 += 4) {
        const int ka = k0 + half * 2;
        v2f a = *(const v2f*)(arow + ka);                 // global_load_b64
        const float* b0 = B + (size_t)ka * HD1 + nb;      // row ka
        const float* b1 = b0 + HD1;                       // row ka+1
        v2f bA; bA.x = b0[0];  bA.y = b1[0];
        v2f bB; bB.x = b0[16]; bB.y = b1[16];
        v2f bC; bC.x = b0[32]; bC.y = b1[32];
        v2f bD; bD.x = b0[48]; bD.y = b1[48];
        acc0 = __builtin_amdgcn_wmma_f32_16x16x4_f32(false, a, false, bA, (short)0, acc0, false, false);
        acc1 = __builtin_amdgcn_wmma_f32_16x16x4_f32(false, a, false, bB, (short)0, acc1, false, false);
        acc2 = __builtin_amdgcn_wmma_f32_16x16x4_f32(false, a, false, bC, (short)0, acc2, false, false);
        acc3 = __builtin_amdgcn_wmma_f32_16x16x4_f32(false, a, false, bD, (short)0, acc3, false, false);
    }

    const int mbase = tM * 16 + half * 8;
    float* drow = D + (size_t)mbase * HD1 + nb;
    #pragma unroll
    for (int v = 0; v < 8; ++v) {
        drow[0]  = acc0[v];
        drow[16] = acc1[v];
        drow[32] = acc2[v];
        drow[48] = acc3[v];
        drow += HD1;
    }
}

// ---------------------------------------------------------------------------
// Generic guarded WMMA GEMM (used for GEMM2, N=22): one wave per 16x16 tile.
// ---------------------------------------------------------------------------
__global__ void wmma_gemm_f32(const float* __restrict__ A,
                              const float* __restrict__ B,
                              float* __restrict__ D,
                              int M, int N, int K, int ntN) {
    int wid  = (blockIdx.x * blockDim.x + threadIdx.x) >> 5;
    int lane = threadIdx.x & 31;
    int tM   = wid / ntN;
    int tN   = wid % ntN;
    if (tM * 16 >= M) return;
    int half = lane >> 4;
    int ln   = lane & 15;
    int m    = tM * 16 + ln;
    int n    = tN * 16 + ln;

    v8f acc = {};
    const float* arow = A + (size_t)m * K;
    const bool inN = (n < N);
    for (int k0 = 0; k0 < K; k0 += 4) {
        int ka = k0 + half * 2;
        v2f a = *(const v2f*)(arow + ka);
        v2f b;
        if (inN) {
            b.x = B[(size_t)ka * N + n];
            b.y = B[(size_t)(ka + 1) * N + n];
        } else {
            b.x = 0.0f; b.y = 0.0f;
        }
        acc = __builtin_amdgcn_wmma_f32_16x16x4_f32(false, a, false, b, (short)0, acc, false, false);
    }
    if (inN) {
        int mbase = tM * 16 + half * 8;
        #pragma unroll
        for (int v = 0; v < 8; ++v)
            D[(size_t)(mbase + v) * N + n] = acc[v];
    }
}

// ---- layer-1 per-(node,head) attention dot products -----------------------
__global__ void dots1_kernel(const float* __restrict__ h,
                             const float* __restrict__ a1,
                             float* __restrict__ dl, float* __restrict__ dr) {
    int t = blockIdx.x * blockDim.x + threadIdx.x;
    if (t >= N_NODES * H1) return;
    int n  = t / H1;
    int hd = t % H1;
    const float* hv = h  + (size_t)n * HD1 + (size_t)hd * D1;
    const float* al = a1 + (size_t)hd * (2 * D1);
    const float* ar = al + D1;
    float sl = 0.0f, sr = 0.0f;
    for (int d = 0; d < D1; ++d) { sl += hv[d] * al[d]; sr += hv[d] * ar[d]; }
    dl[t] = sl;
    dr[t] = sr;
}

// ---- layer-1 edge passes ---------------------------------------------------
__global__ void edge_max1_kernel(const int* __restrict__ ei,
                                 const float* __restrict__ dl,
                                 const float* __restrict__ dr,
                                 float* __restrict__ sbuf,
                                 unsigned* __restrict__ mkey) {
    int t = blockIdx.x * blockDim.x + threadIdx.x;
    if (t >= N_EDGES * H1) return;
    int e  = t / H1;
    int hd = t % H1;
    int r = ei[e];
    int c = ei[N_EDGES + e];
    float s = dl[r * H1 + hd] + dr[c * H1 + hd];
    sbuf[t] = s;
    atomicMax(&mkey[r * H1 + hd], fkey(s));
}

__global__ void edge_exp1_kernel(const int* __restrict__ ei,
                                 const unsigned* __restrict__ mkey,
                                 float* __restrict__ sbuf,
                                 float* __restrict__ z) {
    int t = blockIdx.x * blockDim.x + threadIdx.x;
    if (t >= N_EDGES * H1) return;
    int e  = t / H1;
    int hd = t % H1;
    int r = ei[e];
    float m  = fdec(mkey[r * H1 + hd]);
    float ex = expf(sbuf[t] - m);
    sbuf[t] = ex;
    atomicAdd(&z[r * H1 + hd], ex);
}

// wave per (edge, head): 128-wide message scatter-add (4 atomics / lane)
__global__ void edge_msg1_kernel(const int* __restrict__ ei,
                                 const float* __restrict__ h,
                                 const float* __restrict__ sbuf,
                                 const float* __restrict__ z,
                                 float* __restrict__ agg) {
    int wid  = (blockIdx.x * blockDim.x + threadIdx.x) >> 5;
    int lane = threadIdx.x & 31;
    if (wid >= N_EDGES * H1) return;
    int e  = wid / H1;
    int hd = wid % H1;
    int r = ei[e];
    int c = ei[N_EDGES + e];
    float alpha = sbuf[wid] / z[r * H1 + hd];
    const float* hv = h   + (size_t)r * HD1 + (size_t)hd * D1;
    float*       av = agg + (size_t)c * HD1 + (size_t)hd * D1;
    #pragma unroll
    for (int j = 0; j < 4; ++j) {
        int d = lane + 32 * j;
        atomicAdd(&av[d], alpha * hv[d]);
    }
}

// ---- ELU + mean over heads -------------------------------------------------
__global__ void elu_mean_kernel(const float* __restrict__ agg,
                                float* __restrict__ h1) {
    int t = blockIdx.x * blockDim.x + threadIdx.x;
    if (t >= N_NODES * D1) return;
    int n = t / D1;
    int d = t % D1;
    float s = 0.0f;
    #pragma unroll
    for (int hd = 0; hd < H1; ++hd) {
        float v = agg[(size_t)n * HD1 + (size_t)hd * D1 + d];
        s += (v > 0.0f) ? v : expm1f(v);
    }
    h1[t] = 0.25f * s;
}

// ---- layer-2 per-node dots -------------------------------------------------
__global__ void dots2_kernel(const float* __restrict__ t2,
                             const float* __restrict__ a2,
                             float* __restrict__ dl, float* __restrict__ dr) {
    int n = blockIdx.x * blockDim.x + threadIdx.x;
    if (n >= N_NODES) return;
    float sl = 0.0f, sr = 0.0f;
    const float* tv = t2 + (size_t)n * C2;
    for (int d = 0; d < C2; ++d) { sl += tv[d] * a2[d]; sr += tv[d] * a2[C2 + d]; }
    dl[n] = sl;
    dr[n] = sr;
}

__global__ void edge_max2_kernel(const int* __restrict__ ei,
                                 const float* __restrict__ dl,
                                 const float* __restrict__ dr,
                                 float* __restrict__ sbuf,
                                 unsigned* __restrict__ mkey) {
    int e = blockIdx.x * blockDim.x + threadIdx.x;
    if (e >= N_EDGES) return;
    int r = ei[e];
    int c = ei[N_EDGES + e];
    float s = dl[r] + dr[c];
    sbuf[e] = s;
    atomicMax(&mkey[r], fkey(s));
}

__global__ void edge_exp2_kernel(const int* __restrict__ ei,
                                 const unsigned* __restrict__ mkey,
                                 float* __restrict__ sbuf,
                                 float* __restrict__ z) {
    int e = blockIdx.x * blockDim.x + threadIdx.x;
    if (e >= N_EDGES) return;
    int r = ei[e];
    float ex = expf(sbuf[e] - fdec(mkey[r]));
    sbuf[e] = ex;
    atomicAdd(&z[r], ex);
}

// wave per edge: lanes 0..21 scatter one output feature each
__global__ void edge_msg2_kernel(const int* __restrict__ ei,
                                 const float* __restrict__ t2,
                                 const float* __restrict__ sbuf,
                                 const float* __restrict__ z,
                                 float* __restrict__ out) {
    int wid  = (blockIdx.x * blockDim.x + threadIdx.x) >> 5;
    int lane = threadIdx.x & 31;
    if (wid >= N_EDGES) return;
    int r = ei[wid];
    int c = ei[N_EDGES + wid];
    float alpha = sbuf[wid] / z[r];
    if (lane < C2)
        atomicAdd(&out[(size_t)c * C2 + lane], alpha * t2[(size_t)r * C2 + lane]);
}

// ---------------------------------------------------------------------------
extern "C" void kernel_launch(void* const* d_in, const int* in_sizes, int n_in,
                              void* d_out, int out_size, void* d_ws, size_t ws_size,
                              hipStream_t stream) {
    const float* x  = (const float*)d_in[0];
    const int*   ei = (const int*)  d_in[1];
    const float* W1 = (const float*)d_in[2];
    const float* a1 = (const float*)d_in[3];
    const float* W2 = (const float*)d_in[4];
    const float* a2 = (const float*)d_in[5];
    float* out = (float*)d_out;

    // ---- workspace layout ----
    char* ws = (char*)d_ws;
    size_t off = 0;
    auto alloc = [&](size_t bytes) { char* p = ws + off; off += (bytes + 255) & ~(size_t)255; return p; };
    float*    h     = (float*)   alloc((size_t)N_NODES * HD1 * 4);   // 20.5 MB
    float*    agg1  = (float*)   alloc((size_t)N_NODES * HD1 * 4);   // 20.5 MB
    float*    h1    = (float*)   alloc((size_t)N_NODES * D1  * 4);   //  5.1 MB
    float*    dl1   = (float*)   alloc((size_t)N_NODES * H1  * 4);
    float*    dr1   = (float*)   alloc((size_t)N_NODES * H1  * 4);
    unsigned* m1    = (unsigned*)alloc((size_t)N_NODES * H1  * 4);
    float*    z1    = (float*)   alloc((size_t)N_NODES * H1  * 4);
    float*    e1    = (float*)   alloc((size_t)N_EDGES * H1  * 4);   //  5.1 MB
    float*    t2    = (float*)   alloc((size_t)N_NODES * C2  * 4);
    float*    dl2   = (float*)   alloc((size_t)N_NODES * 4);
    float*    dr2   = (float*)   alloc((size_t)N_NODES * 4);
    unsigned* m2    = (unsigned*)alloc((size_t)N_NODES * 4);
    float*    z2    = (float*)   alloc((size_t)N_NODES * 4);
    float*    e2    = (float*)   alloc((size_t)N_EDGES * 4);
    (void)ws_size; (void)in_sizes; (void)n_in; (void)out_size;

    // zero-init accumulators (0x00000000 is also fkey of the smallest value)
    hipMemsetAsync(agg1, 0, (size_t)N_NODES * HD1 * 4, stream);
    hipMemsetAsync(m1,   0, (size_t)N_NODES * H1  * 4, stream);
    hipMemsetAsync(z1,   0, (size_t)N_NODES * H1  * 4, stream);
    hipMemsetAsync(m2,   0, (size_t)N_NODES * 4,       stream);
    hipMemsetAsync(z2,   0, (size_t)N_NODES * 4,       stream);
    hipMemsetAsync(out,  0, (size_t)N_NODES * C2 * 4,  stream);

    const int BLK = 256;

    // ---- layer 1 ----
    {   // GEMM1: 625 M-tiles x 8 strips of 64 cols, 1 wave per strip
        int waves  = 625 * 8;
        int blocks = (waves * 32 + BLK - 1) / BLK;   // 625 blocks
        wmma_gemm1_kernel<<<blocks, BLK, 0, stream>>>(x, W1, h);
    }
    dots1_kernel<<<(N_NODES * H1 + BLK - 1) / BLK, BLK, 0, stream>>>(h, a1, dl1, dr1);
    edge_max1_kernel<<<(N_EDGES * H1 + BLK - 1) / BLK, BLK, 0, stream>>>(ei, dl1, dr1, e1, m1);
    edge_exp1_kernel<<<(N_EDGES * H1 + BLK - 1) / BLK, BLK, 0, stream>>>(ei, m1, e1, z1);
    {   // wave per (edge, head)
        long long waves = (long long)N_EDGES * H1;
        int blocks = (int)((waves * 32 + BLK - 1) / BLK);
        edge_msg1_kernel<<<blocks, BLK, 0, stream>>>(ei, h, e1, z1, agg1);
    }
    elu_mean_kernel<<<(N_NODES * D1 + BLK - 1) / BLK, BLK, 0, stream>>>(agg1, h1);

    // ---- layer 2 ----
    {   // GEMM2: (10000x128)@(128x22), 625x2 tiles (N guarded)
        int waves  = 625 * 2;
        int blocks = (waves * 32 + BLK - 1) / BLK;
        wmma_gemm_f32<<<blocks, BLK, 0, stream>>>(h1, W2, t2, N_NODES, C2, F1, 2);
    }
    dots2_kernel<<<(N_NODES + BLK - 1) / BLK, BLK, 0, stream>>>(t2, a2, dl2, dr2);
    edge_max2_kernel<<<(N_EDGES + BLK - 1) / BLK, BLK, 0, stream>>>(ei, dl2, dr2, e2, m2);
    edge_exp2_kernel<<<(N_EDGES + BLK - 1) / BLK, BLK, 0, stream>>>(ei, m2, e2, z2);
    {   // wave per edge
        long long waves = N_EDGES;
        int blocks = (int)((waves * 32 + BLK - 1) / BLK);
        edge_msg2_kernel<<<blocks, BLK, 0, stream>>>(ei, t2, e2, z2, out);
    }
}